// LELoss_14027363188885
// MI455X (gfx1250) — compile-verified
//
#include <hip/hip_runtime.h>

// Problem constants (match reference harness).
#define BB   32
#define NN   16384
#define LL   128
#define CH   2048      // points per LDS chunk
#define NCHUNK (NN / CH)
#define NWAVE 8        // 256 threads / wave32

typedef __attribute__((ext_vector_type(2))) float v2f;
typedef __attribute__((ext_vector_type(8))) float v8f;

#define FLT_BIG 3.4028235e38f

// Branchless insert of (d, id) into ascending top-3 (bd0<=bd1<=bd2).
// Every select hangs off a single compare -> v_cmp + v_cndmask, no saveexec.
__device__ __forceinline__ void insert3(float d, int id,
                                        float& bd0, float& bd1, float& bd2,
                                        int& bi0, int& bi1, int& bi2) {
    // conditionally replace the current worst
    const bool c = d < bd2;
    float nd2 = c ? d : bd2;
    int   ni2 = c ? id : bi2;
    // compare-swap (bd1, nd2)
    const bool s1 = nd2 < bd1;
    float t1d = s1 ? nd2 : bd1;
    float t2d = s1 ? bd1 : nd2;
    int   t1i = s1 ? ni2 : bi1;
    int   t2i = s1 ? bi1 : ni2;
    // compare-swap (bd0, t1d)
    const bool s0 = t1d < bd0;
    float u0d = s0 ? t1d : bd0;
    float u1d = s0 ? bd0 : t1d;
    int   u0i = s0 ? t1i : bi0;
    int   u1i = s0 ? bi0 : t1i;
    bd0 = u0d; bd1 = u1d; bd2 = t2d;
    bi0 = u0i; bi1 = u1i; bi2 = t2i;
}

// Async-DMA one 2048-point chunk (24576 B) into LDS: 256 threads x 6 x B128.
// GLOBAL_LOAD_ASYNC_TO_LDS_B128 (GV mode), tracked by ASYNCcnt.
__device__ __forceinline__ void stage_chunk_async(const float* __restrict__ gsrc,
                                                  const float* ldst, int tid) {
    const char* g = (const char*)gsrc + tid * 16;
    // Flat->LDS aperture mapping keeps the LDS byte offset in addr[31:0].
    unsigned l = (unsigned)(size_t)ldst + (unsigned)(tid * 16);
#pragma unroll
    for (int k = 0; k < 6; ++k) {
        asm volatile("global_load_async_to_lds_b128 %0, %1, off"
                     :: "v"(l + k * 4096u), "v"(g + (size_t)k * 4096)
                     : "memory");
    }
}

__global__ __launch_bounds__(256) void leloss_main(
    const float* __restrict__ pred_dis,   // (B*N,3)
    const float* __restrict__ pre_xyz,    // (B*N,3)
    const float* __restrict__ pre_lm,     // (B*L,3)
    const float* __restrict__ target_lm,  // (B*L,3)
    float* __restrict__ out,              // [0]=loss (written by finish), [1..]=pred_lm
    float* __restrict__ block_sums)       // (B,) partial loss sums
{
    __shared__ __align__(16) float spts[2][CH * 3];  // double-buffered chunks
    __shared__ float partial[NWAVE * 2];

    const int b    = blockIdx.x;
    const int tid  = threadIdx.x;
    const int lane = tid & 31;
    const int wave = tid >> 5;
    const int m    = lane & 15;         // column (point) / A-row within half
    const int hi   = lane >> 4;         // lane half: selects K pair / M+8

    const int lmBase = b * LL + wave * 16;

    // ---- A operand: landmark row [-2x,-2y,-2z,1], K striped per ISA layout:
    // lanes 0-15 hold K=0,1 (VGPR0,1); lanes 16-31 hold K=2,3.
    const float plx = pre_lm[(lmBase + m) * 3 + 0];
    const float ply = pre_lm[(lmBase + m) * 3 + 1];
    const float plz = pre_lm[(lmBase + m) * 3 + 2];
    const float normP = plx * plx + ply * ply + plz * plz;

    v2f A;
    A[0] = hi ? (-2.0f * plz) : (-2.0f * plx);
    A[1] = hi ? 1.0f          : (-2.0f * ply);

    // ---- C operand: C[j] = ||p_(j + 8*hi)||^2  (D/C layout: VGPR j is
    // landmark j for lanes 0-15, landmark j+8 for lanes 16-31).
    v8f C;
#pragma unroll
    for (int j = 0; j < 8; ++j)
        C[j] = __shfl(normP, j + hi * 8, 32);

    // Running top-3 per D-row (8 landmark rows per lane).
    float bd0[8], bd1[8], bd2[8];
    int   bi0[8], bi1[8], bi2[8];
#pragma unroll
    for (int j = 0; j < 8; ++j) {
        bd0[j] = FLT_BIG; bd1[j] = FLT_BIG; bd2[j] = FLT_BIG;
        bi0[j] = 0;       bi1[j] = 0;       bi2[j] = 0;
    }

    const float* xyz_base = pre_xyz + (size_t)b * NN * 3;

    // Prologue: async-stage chunk 0.
    stage_chunk_async(xyz_base, spts[0], tid);

    for (int cc = 0; cc < NCHUNK; ++cc) {
        const int cb = cc * CH;
        const bool more = (cc + 1) < NCHUNK;

        // Overlap: issue DMA for the next chunk into the other buffer.
        if (more)
            stage_chunk_async(xyz_base + (size_t)(cb + CH) * 3,
                              spts[(cc + 1) & 1], tid);

        // Oldest 6 async loads (current chunk) must have landed; the 6 just
        // issued for the next chunk may remain outstanding.
        if (more) asm volatile("s_wait_asynccnt 0x6" ::: "memory");
        else      asm volatile("s_wait_asynccnt 0x0" ::: "memory");
        __syncthreads();

        const float* sp = spts[cc & 1];

        for (int t = 0; t < CH; t += 16) {
            // B operand: point column [x,y,z,||x||^2]; VGPR0 = K0/K2 (x|z),
            // VGPR1 = K1/K3 (y|norm) split across lane halves.
            const float px = sp[(t + m) * 3 + 0];
            const float py = sp[(t + m) * 3 + 1];
            const float pz = sp[(t + m) * 3 + 2];
            const float nq = px * px + py * py + pz * pz;
            v2f Bv;
            Bv[0] = hi ? pz : px;
            Bv[1] = hi ? nq : py;

            // D[m][n] = ||p_m||^2 - 2 p.x + ||x_n||^2  (exact squared dists)
            v8f D = __builtin_amdgcn_wmma_f32_16x16x4_f32(
                false, A, false, Bv, (short)0, C, false, false);

            // Wave-uniform skip: steady-state almost no candidate beats bd2.
            bool upd = false;
#pragma unroll
            for (int j = 0; j < 8; ++j)
                upd = upd || (D[j] < bd2[j]);

            if (__ballot(upd) != 0ull) {
                const int pidx = cb + t + m;   // this lane's point column index
#pragma unroll
                for (int j = 0; j < 8; ++j)
                    insert3(D[j], pidx, bd0[j], bd1[j], bd2[j],
                            bi0[j], bi1[j], bi2[j]);
            }
        }
        __syncthreads();   // all waves done with this buffer before reuse
    }

    // ---- Butterfly merge of per-lane top-3 within each 16-lane half.
#pragma unroll
    for (int j = 0; j < 8; ++j) {
        for (int mask = 1; mask < 16; mask <<= 1) {
            float o0 = __shfl_xor(bd0[j], mask, 32);
            float o1 = __shfl_xor(bd1[j], mask, 32);
            float o2 = __shfl_xor(bd2[j], mask, 32);
            int   q0 = __shfl_xor(bi0[j], mask, 32);
            int   q1 = __shfl_xor(bi1[j], mask, 32);
            int   q2 = __shfl_xor(bi2[j], mask, 32);
            insert3(o0, q0, bd0[j], bd1[j], bd2[j], bi0[j], bi1[j], bi2[j]);
            insert3(o1, q1, bd0[j], bd1[j], bd2[j], bi0[j], bi1[j], bi2[j]);
            insert3(o2, q2, bd0[j], bd1[j], bd2[j], bi0[j], bi1[j], bi2[j]);
        }
    }

    // ---- Finalize: lane 0 handles landmarks j (rows 0-7), lane 16 handles j+8.
    float my_sum = 0.0f;
    if (m == 0) {
#pragma unroll
        for (int j = 0; j < 8; ++j) {
            const int lm = lmBase + hi * 8 + j;

            const float w0r = 1.0f / (bd0[j] + 1e-8f);
            const float w1r = 1.0f / (bd1[j] + 1e-8f);
            const float w2r = 1.0f / (bd2[j] + 1e-8f);
            const float inv = 1.0f / (w0r + w1r + w2r);
            const float w0 = w0r * inv, w1 = w1r * inv, w2 = w2r * inv;

            const float* g0 = pred_dis + ((size_t)b * NN + bi0[j]) * 3;
            const float* g1 = pred_dis + ((size_t)b * NN + bi1[j]) * 3;
            const float* g2 = pred_dis + ((size_t)b * NN + bi2[j]) * 3;

            const float lx = pre_lm[lm * 3 + 0] + w0 * g0[0] + w1 * g1[0] + w2 * g2[0];
            const float ly = pre_lm[lm * 3 + 1] + w0 * g0[1] + w1 * g1[1] + w2 * g2[1];
            const float lz = pre_lm[lm * 3 + 2] + w0 * g0[2] + w1 * g1[2] + w2 * g2[2];

            out[1 + lm * 3 + 0] = lx;
            out[1 + lm * 3 + 1] = ly;
            out[1 + lm * 3 + 2] = lz;

            const float ex = target_lm[lm * 3 + 0] - lx;
            const float ey = target_lm[lm * 3 + 1] - ly;
            const float ez = target_lm[lm * 3 + 2] - lz;
            my_sum += ex * ex + ey * ey + ez * ez;
        }
        partial[wave * 2 + hi] = my_sum;
    }
    __syncthreads();

    // Deterministic in-block reduction (fixed order).
    if (tid == 0) {
        float s = 0.0f;
        for (int i = 0; i < NWAVE * 2; ++i) s += partial[i];
        block_sums[b] = s;
    }
}

// Deterministic final reduction: fixed-order sum over 32 block partials.
__global__ void leloss_finish(const float* __restrict__ block_sums,
                              float* __restrict__ out)
{
    if (blockIdx.x == 0 && threadIdx.x == 0) {
        float s = 0.0f;
        for (int i = 0; i < BB; ++i) s += block_sums[i];
        out[0] = s * (1.0f / (float)(BB * LL));
    }
}

extern "C" void kernel_launch(void* const* d_in, const int* in_sizes, int n_in,
                              void* d_out, int out_size, void* d_ws, size_t ws_size,
                              hipStream_t stream) {
    const float* pred_dis  = (const float*)d_in[0];
    const float* pre_xyz   = (const float*)d_in[1];
    const float* pre_lm    = (const float*)d_in[2];
    const float* target_lm = (const float*)d_in[3];
    // d_in[4] = batch ids: segments are equal-size and sorted; implicit here.
    float* out = (float*)d_out;
    float* block_sums = (float*)d_ws;

    leloss_main<<<BB, 256, 0, stream>>>(pred_dis, pre_xyz, pre_lm, target_lm,
                                        out, block_sums);
    leloss_finish<<<1, 32, 0, stream>>>(block_sums, out);
}